// DenseSparseGAT_45097156608488
// MI455X (gfx1250) — compile-verified
//
#include <hip/hip_runtime.h>
#include <math.h>

typedef float v2f __attribute__((ext_vector_type(2)));
typedef float v8f __attribute__((ext_vector_type(8)));

#define FIN   128   // concat feature dim
#define HEADS0  8
#define FOUT0  16
#define COLS0 (HEADS0 * FOUT0)   // 128 output columns of layer-0 projection

// ---------------------------------------------------------------------------
// Generic fill
// ---------------------------------------------------------------------------
__global__ void k_fill(float* __restrict__ p, float v, int n) {
    int i = blockIdx.x * blockDim.x + threadIdx.x;
    if (i < n) p[i] = v;
}

// ---------------------------------------------------------------------------
// Layer-0 projection: hp0[N,128] = concat(dyn0,dyn1,static) @ W0
// W0 is [8,128,16]; flattened output column c = head*16 + o, so
// B(k, c) = w0[head][k][o] = w0_flat[head*2048 + k*16 + o].
// One wave computes a 16-row x 128-col tile with V_WMMA_F32_16X16X4_F32.
//
// Documented A (16x4 f32) layout: lane l holds A[M = l&15, K = 2*(l>>4)+j]
// in component j. Assumed mirrored B (4x16 f32) layout: lane l holds
// B[K = 2*(l>>4)+j, N = l&15] in component j.
// C/D 16x16 f32 layout: vgpr v, lane l -> M = (l>>4)*8 + v, N = l&15.
//
// Tail rows are CLAMPED (not predicated): a WMMA output row M depends only on
// A row M, so garbage A rows only produce garbage D rows, and those stores
// are guarded. This keeps EXEC fully set with no per-iteration branching.
// ---------------------------------------------------------------------------
__global__ __launch_bounds__(256) void k_gemm0(
    const float* __restrict__ dyn0, const float* __restrict__ dyn1,
    const float* __restrict__ stat, const float* __restrict__ w0,
    float* __restrict__ hp0, int nNodes)
{
    __shared__ float sw[FIN * COLS0];            // 64 KB: full W0 tile
    for (int i = threadIdx.x; i < FIN * COLS0; i += 256) sw[i] = w0[i];
    __syncthreads();

    const int lane = threadIdx.x & 31;
    const int wave = threadIdx.x >> 5;
    const int row0 = blockIdx.x * 128 + wave * 16;   // 8 waves * 16 rows
    const int m    = lane & 15;                       // A row / B col in tile
    const int half = lane >> 4;                       // K-pair selector

    int arow = row0 + m;
    if (arow >= nNodes) arow = nNodes - 1;            // clamp, store is guarded

    // region base pointers (concat = [dyn0(32) | dyn1(32) | static(64)])
    const float* __restrict__ aps[3];
    aps[0] = dyn0 + (size_t)arow * 32;
    aps[1] = dyn1 + (size_t)arow * 32;
    aps[2] = stat + (size_t)arow * 64;

    v8f acc[8] = {};

    int kt = 0;
    #pragma unroll
    for (int r = 0; r < 3; ++r) {
        const int nkt = (r == 2) ? 16 : 8;            // 64 static cols
        const float* __restrict__ ap = aps[r];
        #pragma unroll 4
        for (int i = 0; i < nkt; ++i, ++kt) {
            // ---- A fragment: two contiguous row elements (8B aligned)
            const int kloc = i * 4 + half * 2;
            const v2f a = *(const v2f*)(ap + kloc);

            // ---- preload all 8 B fragments, then batch the WMMAs
            const int kk = kt * 4 + half * 2;
            v2f b[8];
            #pragma unroll
            for (int ct = 0; ct < 8; ++ct) {
                b[ct].x = sw[ct * 2048 + (kk + 0) * 16 + m];
                b[ct].y = sw[ct * 2048 + (kk + 1) * 16 + m];
            }
            #pragma unroll
            for (int ct = 0; ct < 8; ++ct) {
                acc[ct] = __builtin_amdgcn_wmma_f32_16x16x4_f32(
                    false, a, false, b[ct], (short)0, acc[ct], false, false);
            }
        }
    }

    // ---- store 16x128 tile
    #pragma unroll
    for (int ct = 0; ct < 8; ++ct) {
        #pragma unroll
        for (int v = 0; v < 8; ++v) {
            const int row = row0 + half * 8 + v;
            if (row < nNodes)
                hp0[row * 128 + ct * 16 + m] = acc[ct][v];
        }
    }
}

// ---------------------------------------------------------------------------
// Per-node attention scores: s0[n,h] = hp0[n,h,:] . asrc0[h,:]; same for t0.
// ---------------------------------------------------------------------------
__global__ void k_scores0(const float* __restrict__ hp0,
                          const float* __restrict__ asrc,
                          const float* __restrict__ atrg,
                          float* __restrict__ s0, float* __restrict__ t0,
                          int nNodes)
{
    int tid = blockIdx.x * blockDim.x + threadIdx.x;
    if (tid >= nNodes * HEADS0) return;
    const int n = tid >> 3, h = tid & 7;
    const float* hv = hp0 + n * 128 + h * 16;
    const float* av = asrc + h * 16;
    const float* bv = atrg + h * 16;
    float s = 0.f, t = 0.f;
    #pragma unroll
    for (int o = 0; o < 16; ++o) {
        const float x = hv[o];
        s += x * av[o];
        t += x * bv[o];
    }
    s0[tid] = s;
    t0[tid] = t;
}

__device__ __forceinline__ float leaky02(float x) {
    return x > 0.f ? x : 0.2f * x;
}

// ---------------------------------------------------------------------------
// Layer-0 edge pass 1: denom0[trg,h] += exp(leaky(s0[src,h] + t0[trg,h]))
// (global-max subtraction cancels in alpha; safely skipped in f32 here)
// ---------------------------------------------------------------------------
__global__ void k_edge0_denom(const int* __restrict__ src,
                              const int* __restrict__ trg,
                              const float* __restrict__ s0,
                              const float* __restrict__ t0,
                              float* __restrict__ denom0, int nEdges)
{
    int tid = blockIdx.x * blockDim.x + threadIdx.x;
    if (tid >= nEdges * HEADS0) return;
    const int e = tid >> 3, h = tid & 7;
    const int sn = src[e], tn = trg[e];
    const float w = expf(leaky02(s0[sn * 8 + h] + t0[tn * 8 + h]));
    atomicAdd(&denom0[tn * 8 + h], w);
}

// ---------------------------------------------------------------------------
// Layer-0 edge pass 2: out0[trg,c] += hp0[src,c] * alpha(e, head(c))
// One thread per (edge, column); exp recomputed (cheap, avoids 25.6MB buffer).
// All hot arrays (s0/t0/denom0/hp0/out0 ~ 60MB) are L2-resident (192MB L2).
// ---------------------------------------------------------------------------
__global__ void k_edge0_scatter(const int* __restrict__ src,
                                const int* __restrict__ trg,
                                const float* __restrict__ s0,
                                const float* __restrict__ t0,
                                const float* __restrict__ denom0,
                                const float* __restrict__ hp0,
                                float* __restrict__ out0, int nEdges)
{
    int tid = blockIdx.x * blockDim.x + threadIdx.x;
    if (tid >= nEdges * 128) return;
    const int e = tid >> 7, c = tid & 127, h = c >> 4;
    const int sn = src[e], tn = trg[e];
    const float w = expf(leaky02(s0[sn * 8 + h] + t0[tn * 8 + h]));
    const float alpha = w / denom0[tn * 8 + h];
    atomicAdd(&out0[tn * 128 + c], hp0[sn * 128 + c] * alpha);
}

// ---------------------------------------------------------------------------
// Layer-1 projection via WMMA: hp1[N,2] = ELU(out0)[N,128] @ w1[128,2]
// w1 is zero-padded to 16 columns in LDS; ELU is applied to the A fragment
// in VALU before the WMMA (exact). One wave = 16 rows, 32 K-steps, 1 tile.
// ---------------------------------------------------------------------------
__global__ __launch_bounds__(256) void k_proj1(
    const float* __restrict__ out0, const float* __restrict__ w1,
    float* __restrict__ hp1, int nNodes)
{
    __shared__ float sw1[FIN * 16];               // 8 KB, cols 2..15 = 0
    for (int i = threadIdx.x; i < FIN * 16; i += 256) {
        const int k = i >> 4, c = i & 15;
        sw1[i] = (c < 2) ? w1[k * 2 + c] : 0.0f;
    }
    __syncthreads();

    const int lane = threadIdx.x & 31;
    const int wave = threadIdx.x >> 5;
    const int row0 = blockIdx.x * 128 + wave * 16;
    const int m    = lane & 15;
    const int half = lane >> 4;

    int arow = row0 + m;
    if (arow >= nNodes) arow = nNodes - 1;        // clamp, store is guarded
    const float* __restrict__ ap = out0 + (size_t)arow * 128;

    v8f acc = {};
    #pragma unroll 4
    for (int kt = 0; kt < FIN / 4; ++kt) {
        const int kk = kt * 4 + half * 2;
        v2f a = *(const v2f*)(ap + kk);
        a.x = a.x > 0.f ? a.x : (expf(a.x) - 1.f);   // ELU(alpha=1)
        a.y = a.y > 0.f ? a.y : (expf(a.y) - 1.f);
        v2f b;
        b.x = sw1[(kk + 0) * 16 + m];
        b.y = sw1[(kk + 1) * 16 + m];
        acc = __builtin_amdgcn_wmma_f32_16x16x4_f32(
            false, a, false, b, (short)0, acc, false, false);
    }

    // only columns 0,1 are real
    if (m < 2) {
        #pragma unroll
        for (int v = 0; v < 8; ++v) {
            const int row = row0 + half * 8 + v;
            if (row < nNodes) hp1[row * 2 + m] = acc[v];
        }
    }
}

// ---------------------------------------------------------------------------
// Layer-1 scores from hp1 (asrc1/atrg1 are [1,2,1])
// ---------------------------------------------------------------------------
__global__ void k_scores1(const float* __restrict__ hp1,
                          const float* __restrict__ asrc1,
                          const float* __restrict__ atrg1,
                          float* __restrict__ s1, float* __restrict__ t1,
                          int nNodes)
{
    int n = blockIdx.x * blockDim.x + threadIdx.x;
    if (n >= nNodes) return;
    const float p0 = hp1[n * 2 + 0], p1 = hp1[n * 2 + 1];
    s1[n] = p0 * asrc1[0] + p1 * asrc1[1];
    t1[n] = p0 * atrg1[0] + p1 * atrg1[1];
}

__global__ void k_edge1_denom(const int* __restrict__ src,
                              const int* __restrict__ trg,
                              const float* __restrict__ s1,
                              const float* __restrict__ t1,
                              float* __restrict__ denom1, int nEdges)
{
    int e = blockIdx.x * blockDim.x + threadIdx.x;
    if (e >= nEdges) return;
    const float w = expf(leaky02(s1[src[e]] + t1[trg[e]]));
    atomicAdd(&denom1[trg[e]], w);
}

__global__ void k_edge1_scatter(const int* __restrict__ src,
                                const int* __restrict__ trg,
                                const float* __restrict__ s1,
                                const float* __restrict__ t1,
                                const float* __restrict__ denom1,
                                const float* __restrict__ hp1,
                                float* __restrict__ out1, int nEdges)
{
    int e = blockIdx.x * blockDim.x + threadIdx.x;
    if (e >= nEdges) return;
    const int sn = src[e], tn = trg[e];
    const float w = expf(leaky02(s1[sn] + t1[tn]));
    const float alpha = w / denom1[tn];
    atomicAdd(&out1[tn * 2 + 0], hp1[sn * 2 + 0] * alpha);
    atomicAdd(&out1[tn * 2 + 1], hp1[sn * 2 + 1] * alpha);
}

// ---------------------------------------------------------------------------
// Final: H=1 so mean over heads is identity; stable 2-class log-softmax.
// ---------------------------------------------------------------------------
__global__ void k_final(const float* __restrict__ out1,
                        float* __restrict__ dst, int nNodes)
{
    int n = blockIdx.x * blockDim.x + threadIdx.x;
    if (n >= nNodes) return;
    const float x0 = out1[n * 2 + 0], x1 = out1[n * 2 + 1];
    const float mx = fmaxf(x0, x1);
    const float lse = mx + logf(expf(x0 - mx) + expf(x1 - mx));
    dst[n * 2 + 0] = x0 - lse;
    dst[n * 2 + 1] = x1 - lse;
}

// ---------------------------------------------------------------------------
// Host-side orchestration
// ---------------------------------------------------------------------------
extern "C" void kernel_launch(void* const* d_in, const int* in_sizes, int n_in,
                              void* d_out, int out_size, void* d_ws, size_t ws_size,
                              hipStream_t stream)
{
    const float* stat  = (const float*)d_in[0];
    const float* dyn0  = (const float*)d_in[1];
    const float* dyn1  = (const float*)d_in[2];
    const int*   src   = (const int*)  d_in[3];
    const int*   trg   = (const int*)  d_in[4];
    const float* w0    = (const float*)d_in[5];
    const float* asrc0 = (const float*)d_in[6];
    const float* atrg0 = (const float*)d_in[7];
    const float* w1    = (const float*)d_in[8];
    const float* asrc1 = (const float*)d_in[9];
    const float* atrg1 = (const float*)d_in[10];
    float* out = (float*)d_out;

    const int N = in_sizes[0] / 64;   // static_emb is [N,64]
    const int E = in_sizes[3];

    // workspace layout (floats)
    float* ws     = (float*)d_ws;
    float* hp0    = ws;                       // N*128
    float* s0     = hp0    + (size_t)N * 128; // N*8
    float* t0     = s0     + (size_t)N * 8;   // N*8
    float* denom0 = t0     + (size_t)N * 8;   // N*8
    float* out0   = denom0 + (size_t)N * 8;   // N*128
    float* hp1    = out0   + (size_t)N * 128; // N*2
    float* s1     = hp1    + (size_t)N * 2;   // N
    float* t1     = s1     + (size_t)N;       // N
    float* denom1 = t1     + (size_t)N;       // N
    float* out1   = denom1 + (size_t)N;       // N*2
    (void)ws_size; (void)n_in; (void)out_size;

    const int B = 256;
    // init accumulators (denominators start at the reference's +1e-16)
    k_fill<<<(N * 8   + B - 1) / B, B, 0, stream>>>(denom0, 1e-16f, N * 8);
    k_fill<<<(N * 128 + B - 1) / B, B, 0, stream>>>(out0,   0.0f,   N * 128);
    k_fill<<<(N       + B - 1) / B, B, 0, stream>>>(denom1, 1e-16f, N);
    k_fill<<<(N * 2   + B - 1) / B, B, 0, stream>>>(out1,   0.0f,   N * 2);

    // layer 0
    k_gemm0<<<(N + 127) / 128, B, 0, stream>>>(dyn0, dyn1, stat, w0, hp0, N);
    k_scores0<<<(N * 8 + B - 1) / B, B, 0, stream>>>(hp0, asrc0, atrg0, s0, t0, N);
    k_edge0_denom<<<(E * 8 + B - 1) / B, B, 0, stream>>>(src, trg, s0, t0, denom0, E);
    k_edge0_scatter<<<(int)(((long long)E * 128 + B - 1) / B), B, 0, stream>>>(
        src, trg, s0, t0, denom0, hp0, out0, E);

    // layer 1
    k_proj1<<<(N + 127) / 128, B, 0, stream>>>(out0, w1, hp1, N);
    k_scores1<<<(N + B - 1) / B, B, 0, stream>>>(hp1, asrc1, atrg1, s1, t1, N);
    k_edge1_denom<<<(E + B - 1) / B, B, 0, stream>>>(src, trg, s1, t1, denom1, E);
    k_edge1_scatter<<<(E + B - 1) / B, B, 0, stream>>>(
        src, trg, s1, t1, denom1, hp1, out1, E);

    // output
    k_final<<<(N + B - 1) / B, B, 0, stream>>>(out1, out, N);
}